// ChamferLoss_72258529788766
// MI455X (gfx1250) — compile-verified
//
#include <hip/hip_runtime.h>
#include <math.h>

typedef __attribute__((ext_vector_type(2))) float v2f;
typedef __attribute__((ext_vector_type(4))) float v4f;
typedef __attribute__((ext_vector_type(8))) float v8f;

// ---------------------------------------------------------------------------
// Pack (x, y, z) -> (x, y, z, x^2+y^2+z^2) for both point clouds.
// ---------------------------------------------------------------------------
__global__ void chamfer_pack_kernel(const float* __restrict__ t,
                                    const float* __restrict__ s,
                                    v4f* __restrict__ pkT,
                                    v4f* __restrict__ pkS, int n) {
  int i = blockIdx.x * blockDim.x + threadIdx.x;
  if (i >= n) return;
  float x = t[3 * i], y = t[3 * i + 1], z = t[3 * i + 2];
  pkT[i] = (v4f){x, y, z, x * x + y * y + z * z};
  x = s[3 * i]; y = s[3 * i + 1]; z = s[3 * i + 2];
  pkS[i] = (v4f){x, y, z, x * x + y * y + z * z};
}

// ---------------------------------------------------------------------------
// Each wave: one 16-row tile of P vs ALL columns of Q via WMMA f32 16x16x4.
//   A(M,K) = (-2px, -2py, -2pz, |p|^2)   B(K,N) = (qx, qy, qz, 1)
//   C(M,N) = |q_N|^2  (per-lane splat)   =>  D = ||p_M - q_N||^2 directly.
// Row-min kept in 8 regs, xor-shuffle min over 16-lane halves, sqrt, store.
// Grid first half: rows = target, cols = state_x (dist1).
// Grid second half: swapped (dist2).
// ---------------------------------------------------------------------------
__global__ void __launch_bounds__(256)
chamfer_rowmin_kernel(const v4f* __restrict__ pkT, const v4f* __restrict__ pkS,
                      float* __restrict__ sq, int n, int blocksPerDir) {
  const int dir  = (blockIdx.x >= (unsigned)blocksPerDir) ? 1 : 0;
  const int blk  = blockIdx.x - dir * blocksPerDir;
  const int wave = threadIdx.x >> 5;
  const int lane = threadIdx.x & 31;
  const int m    = lane & 15;
  const bool lo  = (lane < 16);
  const int row0 = (blk * 8 + wave) * 16;

  const v4f* __restrict__ P = dir ? pkS : pkT;
  const v4f* __restrict__ Q = dir ? pkT : pkS;
  float* __restrict__ out   = sq + dir * n + row0;

  // A operand: lanes 0-15 carry (K0,K1) of row m; lanes 16-31 carry (K2,K3).
  v4f p = P[row0 + m];
  v2f a;
  a.x = lo ? (-2.0f * p.x) : (-2.0f * p.z);
  a.y = lo ? (-2.0f * p.y) : p.w;

  v8f rm;
#pragma unroll
  for (int r = 0; r < 8; ++r) rm[r] = __builtin_inff();

  const int ntiles = n >> 4;
  for (int cb = 0; cb < ntiles; ++cb) {
    v4f q = Q[(cb << 4) + m];          // 16 pts broadcast across both halves
    v2f b;
    b.x = lo ? q.x : q.z;              // B: (K0,K1) low half, (K2,K3) high
    b.y = lo ? q.y : 1.0f;
    v8f c;
#pragma unroll
    for (int r = 0; r < 8; ++r) c[r] = q.w;  // + |q_N|^2 via accumulator

    v8f d = __builtin_amdgcn_wmma_f32_16x16x4_f32(
        /*neg_a=*/false, a, /*neg_b=*/false, b,
        /*c_mod=*/(short)0, c, /*reuse_a=*/false, /*reuse_b=*/false);

#pragma unroll
    for (int r = 0; r < 8; ++r) rm[r] = fminf(rm[r], d[r]);
  }

  // Clamp tiny negative cancellation, then min across 16 lanes of each half.
#pragma unroll
  for (int r = 0; r < 8; ++r) rm[r] = fmaxf(rm[r], 0.0f);
#pragma unroll
  for (int off = 8; off >= 1; off >>= 1) {
#pragma unroll
    for (int r = 0; r < 8; ++r) {
      float o = __shfl_xor(rm[r], off, 32);
      rm[r] = fminf(rm[r], o);
    }
  }
  // C/D layout: VGPR r holds row r (lanes 0-15) and row 8+r (lanes 16-31).
  if (lane == 0) {
#pragma unroll
    for (int r = 0; r < 8; ++r) out[r] = sqrtf(rm[r]);
  } else if (lane == 16) {
#pragma unroll
    for (int r = 0; r < 8; ++r) out[8 + r] = sqrtf(rm[r]);
  }
}

// ---------------------------------------------------------------------------
// Deterministic single-block reduction: loss = scale * sum(sq[0..total)).
// ---------------------------------------------------------------------------
__global__ void chamfer_reduce_kernel(const float* __restrict__ sq,
                                      float* __restrict__ out, int total,
                                      float scale) {
  __shared__ float smem[256];
  float s = 0.0f;
  for (int i = threadIdx.x; i < total; i += 256) s += sq[i];
  smem[threadIdx.x] = s;
  __syncthreads();
  for (int st = 128; st > 0; st >>= 1) {
    if ((int)threadIdx.x < st) smem[threadIdx.x] += smem[threadIdx.x + st];
    __syncthreads();
  }
  if (threadIdx.x == 0) out[0] = smem[0] * scale;
}

extern "C" void kernel_launch(void* const* d_in, const int* in_sizes, int n_in,
                              void* d_out, int out_size, void* d_ws,
                              size_t ws_size, hipStream_t stream) {
  (void)n_in; (void)out_size; (void)ws_size;
  const float* state_x = (const float*)d_in[0];
  const float* target  = (const float*)d_in[1];
  const int n = in_sizes[0] / 3;  // 16384

  char* ws = (char*)d_ws;
  v4f*   pkT = (v4f*)ws;                                // n * 16 bytes
  v4f*   pkS = (v4f*)(ws + (size_t)n * 16);             // n * 16 bytes
  float* sqd = (float*)(ws + (size_t)n * 32);           // 2n * 4 bytes

  chamfer_pack_kernel<<<(n + 255) / 256, 256, 0, stream>>>(target, state_x,
                                                           pkT, pkS, n);

  const int blocksPerDir = n / 128;  // 8 waves/block * 16 rows/wave
  chamfer_rowmin_kernel<<<blocksPerDir * 2, 256, 0, stream>>>(pkT, pkS, sqd, n,
                                                              blocksPerDir);

  // loss = 0.5 * (mean(sqrt d1) + mean(sqrt d2)) * 10 = 5 * sum / n
  chamfer_reduce_kernel<<<1, 256, 0, stream>>>(sqd, (float*)d_out, 2 * n,
                                               5.0f / (float)n);
}